// BlockQuantizer_45449343926973
// MI455X (gfx1250) — compile-verified
//
#include <hip/hip_runtime.h>
#include <stdint.h>

// ---------------------------------------------------------------------------
// Block-quantize (EBIT=8, BITS=8) for MI455X / gfx1250.
// Memory-bound: ~1.5 GB traffic -> ~66us at 23.3 TB/s. Pass 2 uses the CDNA5
// Tensor Data Mover (tensor_load_to_lds + TENSORcnt) for deep async staging.
// ---------------------------------------------------------------------------

typedef __attribute__((ext_vector_type(4))) unsigned int v4u;
typedef __attribute__((ext_vector_type(8))) unsigned int v8u;
typedef __attribute__((ext_vector_type(4))) float        v4f;   // native vector
                                                                // (HIP float4 is a
                                                                // class -> rejected
                                                                // by NT builtins)

#define TILE_FLOATS 4096                    // 16 KB per tile
#define TILE_BYTES  (TILE_FLOATS * 4)
#define QBLOCK      256                     // 8 wave32 per block
#define RBLOCK      256

// -------------------------- elementwise quantize ---------------------------
__device__ __forceinline__ float push_quant(float x, float scale, float inv) {
  // push magnitudes away from zero (matches jnp.where(x>=0, max(x,eps), min(x,-eps)))
  x = (x >= 0.0f) ? fmaxf(x, 1e-10f) : fminf(x, -1e-10f);
  float q = __builtin_rintf(x * scale);      // v_rndne_f32: round half to even
  q = fminf(fmaxf(q, -128.0f), 127.0f);      // clip to [-2^7, 2^7-1]
  return q * inv;
}

// scale factors from the global abs-max bit pattern (max is normal, >= 1e-10)
__device__ __forceinline__ void make_scales(unsigned mb, float* scale, float* inv) {
  int e = (int)((mb >> 23) & 0xFFu) - 127;   // floor(log2(max)); inf -> 128
  if (e > 127) e = 127;                      // clip (lower clip unreachable: e >= -34)
  *scale = __uint_as_float((unsigned)(133 - e) << 23); // 2^(6-e)
  *inv   = __uint_as_float((unsigned)(e + 121) << 23); // 2^(e-6)
}

// ------------------------------ init scratch -------------------------------
__global__ void bq_init(unsigned* wsMax) {
  if (threadIdx.x == 0) *wsMax = __float_as_uint(1e-10f);
}

// --------------------------- pass 1: abs-max -------------------------------
__global__ void __launch_bounds__(RBLOCK)
bq_reduce(const float* __restrict__ in, unsigned* __restrict__ wsMax, long long n) {
  const long long nv = n >> 2;
  const v4f* v = (const v4f*)in;
  float m = 1e-10f;
  const long long i0 = (long long)blockIdx.x * blockDim.x + threadIdx.x;
  const long long stride = (long long)gridDim.x * blockDim.x;
  for (long long i = i0; i < nv; i += stride) {
    v4f a = v[i];                             // global_load_b128
    m = fmaxf(m, fmaxf(fmaxf(fabsf(a.x), fabsf(a.y)),
                       fmaxf(fabsf(a.z), fabsf(a.w))));
  }
  for (long long i = (nv << 2) + i0; i < n; i += stride)   // scalar tail
    m = fmaxf(m, fabsf(in[i]));

  // wave32 reduction
  #pragma unroll
  for (int off = 16; off > 0; off >>= 1)
    m = fmaxf(m, __shfl_xor(m, off, 32));

  __shared__ float sm[RBLOCK / 32];
  const int lane = threadIdx.x & 31, wid = threadIdx.x >> 5;
  if (lane == 0) sm[wid] = m;
  __syncthreads();
  if (wid == 0) {
    m = (lane < RBLOCK / 32) ? sm[lane] : 1e-10f;
    #pragma unroll
    for (int off = 4; off > 0; off >>= 1)
      m = fmaxf(m, __shfl_xor(m, off, 32));
    // abs-float bits compare like u32 -> one atomic per block, order-independent
    if (lane == 0) atomicMax(wsMax, __float_as_uint(m));
  }
}

// ------------------- TDM descriptor build + issue (ISA §8) -----------------
__device__ __forceinline__ void tdm_issue(const float* in, long long tile,
                                          unsigned ldsOff) {
  unsigned long long ga =
      (unsigned long long)in + (unsigned long long)tile * TILE_BYTES;
  v4u g0; v8u g1; v4u g2; v4u g3;
  // group0: count=1 (valid), lds_addr, global_addr[56:0], type=2 ("image")
  g0[0] = 1u;
  g0[1] = ldsOff;
  g0[2] = (unsigned)ga;
  g0[3] = (unsigned)((ga >> 32) & 0x1FFFFFFull) | (2u << 30);
  // group1: workgroup_mask=0 (not in cluster), data_size=2 (4B), 1-D tile
  g1[0] = 2u << 16;
  g1[1] = ((unsigned)TILE_FLOATS & 0xFFFFu) << 16;            // tensor_dim0[15:0]
  g1[2] = (((unsigned)TILE_FLOATS >> 16) & 0xFFFFu) | (1u << 16); // dim0 hi, tensor_dim1=1
  g1[3] = ((unsigned)TILE_FLOATS & 0xFFFFu) << 16;            // tile_dim0
  g1[4] = 0u;                                                 // tile_dim1/2 unused
  g1[5] = (unsigned)TILE_FLOATS;                              // tensor_dim0_stride lo
  g1[6] = 0u;
  g1[7] = 0u;
  g2[0] = g2[1] = g2[2] = g2[3] = 0u;                          // dims 2/3 unused
  g3[0] = g3[1] = g3[2] = g3[3] = 0u;
  asm volatile("tensor_load_to_lds %0, %1, %2, %3"
               :: "s"(g0), "s"(g1), "s"(g2), "s"(g3)
               : "memory");
}

// ------------------- pass 2: quantize via TDM double-buffer ----------------
__global__ void __launch_bounds__(QBLOCK)
bq_quant_tdm(const float* __restrict__ in, float* __restrict__ out,
             const unsigned* __restrict__ wsMax, long long nTiles) {
  extern __shared__ float lds[];             // 2 * TILE_BYTES, dynamic -> offset 0

  float scale, inv;
  make_scales(*wsMax, &scale, &inv);

  const long long gstride = gridDim.x;
  long long l = blockIdx.x;
  const bool issuer = (threadIdx.x < 32);    // wave 0 only issues DMA

  // Reverse tile order: reduction's forward sweep leaves high addresses hot
  // in the 192 MB L2 -> read those first instead of LRU-thrashing.
  if (l < nTiles && issuer) tdm_issue(in, nTiles - 1 - l, 0u);

  int buf = 0;
  for (; l < nTiles; l += gstride, buf ^= 1) {
    const long long ln = l + gstride;
    const bool more = (ln < nTiles);
    if (more && issuer)
      tdm_issue(in, nTiles - 1 - ln, (buf ^ 1) ? (unsigned)TILE_BYTES : 0u);
    if (more) __builtin_amdgcn_s_wait_tensorcnt(1);  // current tile landed
    else      __builtin_amdgcn_s_wait_tensorcnt(0);
    __syncthreads();                                 // release all 8 waves

    const long long tile = nTiles - 1 - l;
    const v4f* lbuf = (const v4f*)(lds + (buf ? TILE_FLOATS : 0));
    v4f* op = (v4f*)out + tile * (TILE_FLOATS / 4);
    #pragma unroll 4
    for (int i = threadIdx.x; i < TILE_FLOATS / 4; i += QBLOCK) {
      v4f v = lbuf[i];                               // ds_load_b128
      v4f r;
      r.x = push_quant(v.x, scale, inv);
      r.y = push_quant(v.y, scale, inv);
      r.z = push_quant(v.z, scale, inv);
      r.w = push_quant(v.w, scale, inv);
      __builtin_nontemporal_store(r, op + i);        // global_store_b128 NT
    }
    __syncthreads();                                 // buffer free for reuse
  }
}

// --------------------------- generic tail kernel ---------------------------
__global__ void bq_quant_tail(const float* __restrict__ in, float* __restrict__ out,
                              const unsigned* __restrict__ wsMax,
                              long long start, long long n) {
  float scale, inv;
  make_scales(*wsMax, &scale, &inv);
  long long i = start + (long long)blockIdx.x * blockDim.x + threadIdx.x;
  if (i < start + n) out[i] = push_quant(in[i], scale, inv);
}

// --------------------------------- launch ----------------------------------
static inline long long llmin(long long a, long long b) { return a < b ? a : b; }

extern "C" void kernel_launch(void* const* d_in, const int* in_sizes, int n_in,
                              void* d_out, int out_size, void* d_ws, size_t ws_size,
                              hipStream_t stream) {
  const float* x = (const float*)d_in[0];
  float* out = (float*)d_out;
  unsigned* wsMax = (unsigned*)d_ws;
  const long long n = (long long)in_sizes[0];

  // 0) reset scratch max every call (ws is poisoned, never re-cleared)
  bq_init<<<1, 32, 0, stream>>>(wsMax);

  // 1) global abs-max
  {
    long long nv = (n >> 2) > 0 ? (n >> 2) : n;
    int rgrid = (int)llmin((nv + RBLOCK - 1) / RBLOCK, 4096);
    if (rgrid < 1) rgrid = 1;
    bq_reduce<<<rgrid, RBLOCK, 0, stream>>>(x, wsMax, n);
  }

  // 2) quantize: TDM-staged tiles (double-buffered, 32 KB dynamic LDS)
  const long long nTiles = n / TILE_FLOATS;
  if (nTiles > 0) {
    int qgrid = (int)llmin(nTiles, 4096);
    bq_quant_tdm<<<qgrid, QBLOCK, 2 * TILE_BYTES, stream>>>(x, out, wsMax, nTiles);
  }

  // 3) remainder (none for 16*2048*4096, kept for generality)
  const long long tail0 = nTiles * (long long)TILE_FLOATS;
  if (tail0 < n) {
    long long tn = n - tail0;
    int tgrid = (int)((tn + 255) / 256);
    bq_quant_tail<<<tgrid, 256, 0, stream>>>(x, out, wsMax, tail0, tn);
  }
}